// PQMatcher_27453430956675
// MI455X (gfx1250) — compile-verified
//
#include <hip/hip_runtime.h>
#include <hip/hip_bf16.h>
#include <math.h>

// Problem dims (fixed by the reference)
#define LP 400
#define LQ 50
#define BZ 32
#define SZ 768
#define HH 128

typedef float v2f __attribute__((ext_vector_type(2)));
typedef float v8f __attribute__((ext_vector_type(8)));
typedef int   v4i __attribute__((ext_vector_type(4)));

#define GLOBAL_AS __attribute__((address_space(1)))
#define LDS_AS    __attribute__((address_space(3)))

// -----------------------------------------------------------------------------
// Kernel 1: out[M,128] = A[M,768] * W[128,768]^T + bias[128]  (FP32 WMMA 16x16x4)
// One wave computes a 16(M) x 128(N) strip. K is a compile-time constant so the
// 8 B-tile addresses are (base + nt*16*768*4) immediate offsets -> the backend
// can clause all loads and stagger s_wait_loadcnt instead of wait(0) per WMMA.
// Loads are issued for all 8 N-tiles before the 8 WMMAs to overlap latency.
// A-tile layout (ISA 7.12.2, 32-bit A 16x4): lanes 0-15: M=lane, K0/K0+1;
// lanes 16-31: M=lane-16, K0+2/K0+3. B-tile 4x16 mirrors it with N=lane&15.
// -----------------------------------------------------------------------------
__global__ void pq_proj_gemm(const float* __restrict__ A,
                             const float* __restrict__ W,
                             const float* __restrict__ bias,
                             float* __restrict__ out) {
  const int lane  = threadIdx.x;          // 0..31
  const int m0    = blockIdx.x * 16;
  const int idx   = lane & 15;
  const int khalf = (lane >> 4) * 2;      // 0 for lanes 0-15, 2 for 16-31

  const float* __restrict__ arow = A + (m0 + idx) * SZ + khalf;
  const float* __restrict__ wrow = W + idx * SZ + khalf;

  v8f acc[8];
#pragma unroll
  for (int nt = 0; nt < 8; ++nt) acc[nt] = (v8f){0.f,0.f,0.f,0.f,0.f,0.f,0.f,0.f};

  for (int k0 = 0; k0 < SZ; k0 += 4) {
    v2f a; a.x = arow[k0]; a.y = arow[k0 + 1];
    v2f b[8];
#pragma unroll
    for (int nt = 0; nt < 8; ++nt) {       // constant offsets: nt*16*768 floats
      b[nt].x = wrow[nt * 16 * SZ + k0];
      b[nt].y = wrow[nt * 16 * SZ + k0 + 1];
    }
#pragma unroll
    for (int nt = 0; nt < 8; ++nt)
      acc[nt] = __builtin_amdgcn_wmma_f32_16x16x4_f32(
          false, a, false, b[nt], (short)0, acc[nt], false, false);
  }

  const int rowbase = m0 + ((lane >> 4) << 3);   // +8 for upper half-wave
#pragma unroll
  for (int nt = 0; nt < 8; ++nt) {
    const int col = nt * 16 + idx;
    const float bi = bias[col];
    for (int r = 0; r < 8; ++r)
      out[(rowbase + r) * HH + col] = acc[nt][r] + bi;
  }
}

// -----------------------------------------------------------------------------
// Kernel 2: Wvv[b,g] = sum_h v[b,h] * Wv_w[g,h] + Wv_b[g]    (tiny: 32x128)
// -----------------------------------------------------------------------------
__global__ void pq_wvv(const float* __restrict__ v,
                       const float* __restrict__ Wv_w,
                       const float* __restrict__ Wv_b,
                       float* __restrict__ Wvv) {
  int idx = blockIdx.x * blockDim.x + threadIdx.x;   // 0..4095
  int b = idx >> 7, g = idx & 127;
  float acc = Wv_b[g];
  const float* __restrict__ vp = v    + b * HH;
  const float* __restrict__ wp = Wv_w + g * HH;
  for (int h = 0; h < HH; ++h) acc += vp[h] * wp[h];
  Wvv[idx] = acc;
}

// -----------------------------------------------------------------------------
// Kernel 3: per (p,b): s[q] = sum_h tanh(Wup+Wuq+Wvv) * v ; softmax over q.
// Block = 256 threads (8 waves) = 8 passage positions sharing one batch b.
// The per-batch Wuq slice (50x128 f32 = 25.6KB) is staged into LDS once via
// async global->LDS copies (ASYNCcnt), then each wave loops q reading LDS.
// Scores stay in registers via owner-lane selects (no LDS score buffer).
// -----------------------------------------------------------------------------
__global__ void __launch_bounds__(256)
pq_attn_softmax(const float* __restrict__ Wup,   // [LP*BZ, H]
                const float* __restrict__ Wuq,   // [LQ*BZ, H]
                const float* __restrict__ Wvv,   // [BZ, H]
                const float* __restrict__ v,     // [BZ, H]
                float* __restrict__ a_out) {     // [LP*BZ, LQ]
  const int b     = blockIdx.x;           // batch
  const int pbase = blockIdx.y * 8;       // 8 passage positions per block
  const int tid   = threadIdx.x;
  const int wave  = tid >> 5;
  const int lane  = tid & 31;

  __shared__ float wq_lds[LQ * HH];       // 25.6 KB

  // ---- stage Wuq[:, b, :] into LDS (1600 x 16B chunks over 256 threads) ----
  for (int t = tid; t < (LQ * HH) / 4; t += 256) {
    const int q  = t >> 5;                 // 32 chunks per 128-float row
    const int c4 = t & 31;
    const float* src = Wuq + (q * BZ + b) * HH + c4 * 4;
    float* dst = &wq_lds[t * 4];
#if __has_builtin(__builtin_amdgcn_global_load_async_to_lds_b128)
    __builtin_amdgcn_global_load_async_to_lds_b128(
        (GLOBAL_AS v4i*)src, (LDS_AS v4i*)dst, 0, 0);
#else
    dst[0] = src[0]; dst[1] = src[1]; dst[2] = src[2]; dst[3] = src[3];
#endif
  }
#if __has_builtin(__builtin_amdgcn_global_load_async_to_lds_b128)
#if __has_builtin(__builtin_amdgcn_s_wait_asynccnt)
  __builtin_amdgcn_s_wait_asynccnt(0);
#else
  asm volatile("s_wait_asynccnt 0x0" ::: "memory");
#endif
#endif
  __syncthreads();

  // ---- per-wave score loop ----
  const int p = pbase + wave;
  const int m = p * BZ + b;

  float base[4], vv[4];
#pragma unroll
  for (int j = 0; j < 4; ++j) {
    int h = lane + j * 32;
    base[j] = Wup[m * HH + h] + Wvv[b * HH + h];
    vv[j]   = v[b * HH + h];
  }

  float s1 = -3.4e38f, s2 = -3.4e38f;      // lane owns q=lane and q=lane+32
  for (int q = 0; q < LQ; ++q) {
    float partial = 0.f;
#pragma unroll
    for (int j = 0; j < 4; ++j)
      partial += tanhf(base[j] + wq_lds[q * HH + lane + j * 32]) * vv[j];
    for (int off = 16; off; off >>= 1)
      partial += __shfl_xor(partial, off, 32);
    s1 = (q < 32 && lane == q)        ? partial : s1;
    s2 = (q >= 32 && lane == q - 32)  ? partial : s2;
  }

  // ---- softmax over 50 values ----
  float mx = fmaxf(s1, s2);
  for (int off = 16; off; off >>= 1)
    mx = fmaxf(mx, __shfl_xor(mx, off, 32));
  float e1 = __expf(s1 - mx);
  float e2 = (lane + 32 < LQ) ? __expf(s2 - mx) : 0.f;
  float sum = e1 + e2;
  for (int off = 16; off; off >>= 1)
    sum += __shfl_xor(sum, off, 32);
  float inv = 1.f / sum;
  a_out[m * LQ + lane] = e1 * inv;
  if (lane + 32 < LQ) a_out[m * LQ + lane + 32] = e2 * inv;
}

// -----------------------------------------------------------------------------
// Kernel 4: c[p,b,s] = sum_q a[p,b,q] * Uq[q,b,s]   (per-b GEMM, K=50 pad->52)
// One wave computes a 16(p) x 64(s) strip (4 N-tiles reuse the A fragment).
// Fully unrolled K loop (13 steps, constant strides); tail guard clamps the
// load index (always in-bounds) and selects the value with v_cndmask -> no
// divergence, EXEC stays all-1s for WMMA. Loads issued before the 4 WMMAs.
// -----------------------------------------------------------------------------
__global__ void pq_ctx_gemm(const float* __restrict__ Aa,   // [LP*BZ, LQ]
                            const float* __restrict__ Uq,   // [LQ*BZ, SZ]
                            float* __restrict__ out) {      // [LP*BZ, SZ]
  const int lane  = threadIdx.x;
  const int s0    = blockIdx.x * 64;
  const int p0    = blockIdx.y * 16;
  const int b     = blockIdx.z;
  const int idx   = lane & 15;
  const int khalf = (lane >> 4) * 2;

  const float* __restrict__ arow = Aa + ((p0 + idx) * BZ + b) * LQ;
  const float* __restrict__ ubase = Uq + b * SZ + s0 + idx;

  v8f acc[4];
#pragma unroll
  for (int nt = 0; nt < 4; ++nt) acc[nt] = (v8f){0.f,0.f,0.f,0.f,0.f,0.f,0.f,0.f};

#pragma unroll
  for (int k0 = 0; k0 < 52; k0 += 4) {
    const int ka  = k0 + khalf;
    const int ka0 = (ka     < LQ) ? ka     : (LQ - 1);   // clamp: loads in-bounds
    const int ka1 = (ka + 1 < LQ) ? ka + 1 : (LQ - 1);
    const bool ok0 = (ka     < LQ);
    const bool ok1 = (ka + 1 < LQ);

    float a0 = arow[ka0], a1 = arow[ka1];
    float b0[4], b1[4];
#pragma unroll
    for (int nt = 0; nt < 4; ++nt) {
      b0[nt] = ubase[ka0 * (BZ * SZ) + nt * 16];
      b1[nt] = ubase[ka1 * (BZ * SZ) + nt * 16];
    }
    v2f a; a.x = ok0 ? a0 : 0.f; a.y = ok1 ? a1 : 0.f;
#pragma unroll
    for (int nt = 0; nt < 4; ++nt) {
      v2f bv; bv.x = ok0 ? b0[nt] : 0.f; bv.y = ok1 ? b1[nt] : 0.f;
      acc[nt] = __builtin_amdgcn_wmma_f32_16x16x4_f32(
          false, a, false, bv, (short)0, acc[nt], false, false);
    }
  }

  const int rowbase = p0 + ((lane >> 4) << 3);
#pragma unroll
  for (int nt = 0; nt < 4; ++nt)
    for (int r = 0; r < 8; ++r)
      out[((rowbase + r) * BZ + b) * SZ + s0 + nt * 16 + idx] = acc[nt][r];
}

// -----------------------------------------------------------------------------
extern "C" void kernel_launch(void* const* d_in, const int* in_sizes, int n_in,
                              void* d_out, int out_size, void* d_ws, size_t ws_size,
                              hipStream_t stream) {
  (void)in_sizes; (void)n_in; (void)out_size; (void)ws_size;
  const float* Up   = (const float*)d_in[0];
  const float* Uq   = (const float*)d_in[1];
  const float* Wp_w = (const float*)d_in[2];
  const float* Wp_b = (const float*)d_in[3];
  const float* Wq_w = (const float*)d_in[4];
  const float* Wq_b = (const float*)d_in[5];
  const float* Wv_w = (const float*)d_in[6];
  const float* Wv_b = (const float*)d_in[7];
  const float* v    = (const float*)d_in[8];
  float* out = (float*)d_out;

  // scratch layout (floats)
  float* ws   = (float*)d_ws;
  float* Wup  = ws;                                  // LP*BZ*H = 1,638,400
  float* Wuq  = Wup + (size_t)LP * BZ * HH;          // LQ*BZ*H =   204,800
  float* Wvv  = Wuq + (size_t)LQ * BZ * HH;          // BZ*H    =     4,096
  float* Aatt = Wvv + (size_t)BZ * HH;               // LP*BZ*LQ=   640,000

  // 1) Wup = Up @ Wp_w^T + b    (M = 12800 -> 800 strip-waves)
  pq_proj_gemm<<<(LP * BZ) / 16, 32, 0, stream>>>(Up, Wp_w, Wp_b, Wup);
  // 2) Wuq = Uq @ Wq_w^T + b    (M = 1600)
  pq_proj_gemm<<<(LQ * BZ) / 16, 32, 0, stream>>>(Uq, Wq_w, Wq_b, Wuq);
  // 3) Wvv = v @ Wv_w^T + b
  pq_wvv<<<(BZ * HH) / 256, 256, 0, stream>>>(v, Wv_w, Wv_b, Wvv);
  // 4) scores + softmax -> attention weights (8 waves/block share LDS stage)
  pq_attn_softmax<<<dim3(BZ, LP / 8), 256, 0, stream>>>(Wup, Wuq, Wvv, v, Aatt);
  // 5) contexts: c = a @ Uq (batched over b), 16x64 strips
  pq_ctx_gemm<<<dim3(SZ / 64, LP / 16, BZ), 32, 0, stream>>>(Aatt, Uq, out);
}